// HungarianMatcher_44538810860259
// MI455X (gfx1250) — compile-verified
//
#include <hip/hip_runtime.h>
#include <cstdint>

// Problem shape (fixed by reference): B=64, Q=1000, T=500 (logits/classes unused).
#define B_  64
#define Q_  1000
#define T_  500
#define TP  512     // padded target row length (avoids OOB + clean banking)
#define QT  50      // queries per block -> grid.x = 20

typedef uint32_t u32;
typedef __attribute__((ext_vector_type(4))) u32 u32x4;
typedef __attribute__((ext_vector_type(8))) int i32x8;
typedef __attribute__((ext_vector_type(4))) int i32x4;

// ---------------------------------------------------------------------------
// Tensor Data Mover: 1-D tile of nElem f32 elements, global -> LDS.
// D# per CDNA5 ISA ch.8:
//   group0: [1:0]=count=1, [63:32]=lds_addr, [120:64]=global_addr, [127:126]=type=2
//   group1: [17:16]=data_size=2 (4B), tensor_dim0=nElem, tensor_dim1=1,
//           tile_dim0=nElem (1-D), tile_dim1/2=0, dim0_stride=nElem
//   groups 2/3: zero (<=2-D tensor)
// This toolchain's builtin arity is 6: (u32x4, i32x8, i32x4, i32x4, i32x8, i32 cpol)
// ---------------------------------------------------------------------------
__device__ __forceinline__ void tdm_load_f32_1d(u32 lds_off, const float* gptr, u32 nElem) {
    uint64_t ga = (uint64_t)(uintptr_t)gptr;
    u32x4 g0;
    g0.x = 1u;                                                  // count=1, user mode
    g0.y = lds_off;                                             // lds_addr (bytes)
    g0.z = (u32)(ga & 0xFFFFFFFFull);                           // global_addr[31:0]
    g0.w = ((u32)(ga >> 32) & 0x01FFFFFFu) | 0x80000000u;       // addr[56:32] | type=2
    i32x8 g1;
    g1[0] = (int)(2u << 16);                                    // data_size = 4 bytes
    g1[1] = (int)((nElem & 0xFFFFu) << 16);                     // tensor_dim0[15:0]
    g1[2] = (int)((nElem >> 16) | (1u << 16));                  // dim0[31:16] | tensor_dim1=1
    g1[3] = (int)((nElem & 0xFFFFu) << 16);                     // tile_dim0 = nElem
    g1[4] = 0;                                                  // tile_dim1=0, tile_dim2=0
    g1[5] = (int)nElem;                                         // tensor_dim0_stride lo
    g1[6] = 0;
    g1[7] = 0;
    i32x4 z4 = {0, 0, 0, 0};
    i32x8 z8 = {0, 0, 0, 0, 0, 0, 0, 0};
    __builtin_amdgcn_tensor_load_to_lds(g0, g1, z4, z4, z8, 0);
}

// cost(q,t) = 5*L1(cxcywh) - 2*GIoU(xyxy).  ~32 VALU + 2 v_rcp_f32.
__device__ __forceinline__ float pair_cost(
    float qcx, float qcy, float qw, float qh,
    float qx1, float qy1, float qx2, float qy2, float qa,
    float tcx, float tcy, float tw, float th,
    float tx1, float ty1, float tx2, float ty2, float ta)
{
    float l1 = fabsf(qcx - tcx) + fabsf(qcy - tcy) + fabsf(qw - tw) + fabsf(qh - th);
    float iw = fminf(qx2, tx2) - fmaxf(qx1, tx1);
    float ih = fminf(qy2, ty2) - fmaxf(qy1, ty1);
    float inter = fmaxf(iw, 0.f) * fmaxf(ih, 0.f);
    float uni = qa + ta - inter;
    float cw = fmaxf(qx2, tx2) - fminf(qx1, tx1);
    float chh = fmaxf(qy2, ty2) - fminf(qy1, ty1);
    float ca = fmaxf(cw, 0.f) * fmaxf(chh, 0.f);
    // giou = inter/uni - (ca-uni)/ca ;  cost = 5*l1 - 2*giou
    float giou = inter * __builtin_amdgcn_rcpf(uni)
               - (ca - uni) * __builtin_amdgcn_rcpf(ca);
    return fmaf(-2.f, giou, 5.f * l1);
}

__global__ __launch_bounds__(256)
void HungarianMatcher_44538810860259_kernel(const float* __restrict__ pred_boxes,
                                            const float* __restrict__ target_boxes,
                                            float* __restrict__ out)
{
    __shared__ float s_stage[T_ * 4 + 48];   // raw target boxes via TDM (8 KB)
    __shared__ float s_t[9][TP];             // target SoA: cx,cy,w,h,x1,y1,x2,y2,area
    __shared__ float s_q[9][QT];             // query  SoA

    const int tid = threadIdx.x;
    const int b  = blockIdx.y;
    const int qb = blockIdx.x * QT;

    // --- Stage this batch's 500 target boxes into LDS via the Tensor Data Mover.
    // TDM ignores EXEC and issues once per wave; gate so exactly wave 0 issues,
    // and that same wave drains TENSORcnt before the workgroup barrier.
    if (tid < 32) {
        u32 lds_off = (u32)(uintptr_t)(void*)&s_stage[0];   // addrspace(3) low bits = LDS offset
        tdm_load_f32_1d(lds_off, target_boxes + (size_t)b * (T_ * 4), T_ * 4);
        __builtin_amdgcn_s_wait_tensorcnt(0);
    }
    __syncthreads();

    // --- Derive per-target values once (amortized over this block's 50 queries).
    for (int i = tid; i < T_; i += 256) {
        float cx = s_stage[i * 4 + 0], cy = s_stage[i * 4 + 1];
        float w  = s_stage[i * 4 + 2], h  = s_stage[i * 4 + 3];
        s_t[0][i] = cx;  s_t[1][i] = cy;  s_t[2][i] = w;  s_t[3][i] = h;
        s_t[4][i] = cx - 0.5f * w;  s_t[5][i] = cy - 0.5f * h;
        s_t[6][i] = cx + 0.5f * w;  s_t[7][i] = cy + 0.5f * h;
        s_t[8][i] = w * h;
    }
    // --- Derive per-query values once (tiny: 50 boxes).
    for (int i = tid; i < QT; i += 256) {
        const float* p = pred_boxes + ((size_t)b * Q_ + (qb + i)) * 4;
        float cx = p[0], cy = p[1], w = p[2], h = p[3];
        s_q[0][i] = cx;  s_q[1][i] = cy;  s_q[2][i] = w;  s_q[3][i] = h;
        s_q[4][i] = cx - 0.5f * w;  s_q[5][i] = cy - 0.5f * h;
        s_q[6][i] = cx + 0.5f * w;  s_q[7][i] = cy + 0.5f * h;
        s_q[8][i] = w * h;
    }
    __syncthreads();

    // --- Each thread owns targets t0=tid and t1=tid+256; hoist them to regs.
    const int  t0   = tid;
    const int  t1   = tid + 256;
    const bool has1 = (t1 < T_);

    float a0 = s_t[0][t0], a1 = s_t[1][t0], a2 = s_t[2][t0], a3 = s_t[3][t0];
    float a4 = s_t[4][t0], a5 = s_t[5][t0], a6 = s_t[6][t0], a7 = s_t[7][t0];
    float a8 = s_t[8][t0];
    float c0 = s_t[0][t1], c1 = s_t[1][t1], c2 = s_t[2][t1], c3 = s_t[3][t1];
    float c4 = s_t[4][t1], c5 = s_t[5][t1], c6 = s_t[6][t1], c7 = s_t[7][t1];
    float c8 = s_t[8][t1];

    float* __restrict__ row = out + ((size_t)b * Q_ + qb) * T_;

    #pragma unroll 2
    for (int q = 0; q < QT; ++q) {
        // uniform broadcast loads from LDS
        float q0 = s_q[0][q], q1 = s_q[1][q], q2 = s_q[2][q], q3 = s_q[3][q];
        float q4 = s_q[4][q], q5 = s_q[5][q], q6 = s_q[6][q], q7 = s_q[7][q];
        float q8 = s_q[8][q];

        float r0 = pair_cost(q0,q1,q2,q3,q4,q5,q6,q7,q8,
                             a0,a1,a2,a3,a4,a5,a6,a7,a8);
        __builtin_nontemporal_store(r0, row + t0);          // coalesced, write-once
        if (has1) {
            float r1 = pair_cost(q0,q1,q2,q3,q4,q5,q6,q7,q8,
                                 c0,c1,c2,c3,c4,c5,c6,c7,c8);
            __builtin_nontemporal_store(r1, row + t1);
        }
        row += T_;
    }
}

extern "C" void kernel_launch(void* const* d_in, const int* in_sizes, int n_in,
                              void* d_out, int out_size, void* d_ws, size_t ws_size,
                              hipStream_t stream) {
    // setup_inputs order: [0]=pred_logits (unused), [1]=pred_boxes,
    //                     [2]=target_classes (unused), [3]=target_boxes
    const float* pred_boxes   = (const float*)d_in[1];
    const float* target_boxes = (const float*)d_in[3];
    float* out = (float*)d_out;

    dim3 grid(Q_ / QT, B_);   // (20, 64) = 1280 blocks, each 8 wave32
    dim3 block(256);
    hipLaunchKernelGGL(HungarianMatcher_44538810860259_kernel, grid, block, 0, stream,
                       pred_boxes, target_boxes, out);
}